// GMM_46755013984568
// MI455X (gfx1250) — compile-verified
//
#include <hip/hip_runtime.h>
#include <math.h>

// GMM clamped log-density, hybrid WMMA + VALU split (exact):
//   sum_d min(dist,20) = [bilinear GEMM via V_WMMA_F32_16X16X4_F32]
//                      - [VALU correction sum_d relu(dist-20), 4 ops/elem]
// dist = ((x-mu)/sd)^2, sd = max(std,0.1).  s = 1/sd, c = -mu/sd,
// w = s^2, -2*mu*w = 2*c*s, mu^2*w = c^2 (folded into off2).

typedef __attribute__((ext_vector_type(2))) float v2f;
typedef __attribute__((ext_vector_type(8))) float v8f;

#define GMM_D  32
#define GMM_K  16
#define TPB    256
#define TILE   128      // points per block = 8 waves x 16
#define XSTR   36       // padded dword stride (144B) -> conflict-free b128

__global__ __launch_bounds__(512)
void gmm_prep(const float* __restrict__ mu, const float* __restrict__ stdv,
              const float* __restrict__ alpha,
              float* __restrict__ s, float* __restrict__ c,
              float* __restrict__ off2) {
    const int t = threadIdx.x;                 // 512 threads: one per (k,d)
    if (t < GMM_K * GMM_D) {
        float sd  = fmaxf(stdv[t], 0.1f);
        float inv = 1.0f / sd;
        s[t] = inv;
        c[t] = -mu[t] * inv;
    }
    if (t < GMM_K) {
        float m = -INFINITY;
        for (int i = 0; i < GMM_K; ++i) m = fmaxf(m, alpha[i]);
        float sum = 0.0f;
        for (int i = 0; i < GMM_K; ++i) sum += __expf(alpha[i] - m);
        float log_a = alpha[t] - m - __logf(sum);
        float lacc = 0.0f, q2 = 0.0f;
        for (int d = 0; d < GMM_D; ++d) {
            float sd  = fmaxf(stdv[t * GMM_D + d], 0.1f);
            float inv = 1.0f / sd;
            float cc  = -mu[t * GMM_D + d] * inv;
            lacc -= __logf(sd);
            q2   += cc * cc;                    // sum_d mu^2 * w
        }
        const float HALF_LOG_2PI = 0.91893853320467274178f;
        off2[t] = log_a + lacc - (float)GMM_D * HALF_LOG_2PI - 0.5f * q2;
    }
}

__global__ __launch_bounds__(TPB)
void gmm_main(const float* __restrict__ x,
              const float* __restrict__ s, const float* __restrict__ c,
              const float* __restrict__ off2,
              float* __restrict__ out, int N) {
    __shared__ float xs[TILE * XSTR];          // 18 KB staged x tile
    __shared__ float sP[GMM_K * XSTR];         // padded [k][36]
    __shared__ float cP[GMM_K * XSTR];
    __shared__ float off2S[GMM_K];

    const int t        = threadIdx.x;
    const int tileBase = blockIdx.x * TILE;

    // Parameter block -> LDS (padded stride)
    for (int idx = t; idx < GMM_K * GMM_D; idx += TPB) {
        int kk = idx >> 5, dd = idx & 31;
        sP[kk * XSTR + dd] = s[idx];
        cP[kk * XSTR + dd] = c[idx];
    }
    if (t < GMM_K) off2S[t] = off2[t];

    // ---- Async DMA x tile (128 points x 128B = 1024 x 16B chunks) ---------
    {
        const size_t totalElems = (size_t)N * GMM_D;
        #pragma unroll
        for (int j = 0; j < 4; ++j) {
            int i = t + j * TPB;
            int n = i >> 3, q = i & 7;
            unsigned lds_off = (unsigned)(uintptr_t)(&xs[0])
                             + (unsigned)(n * (XSTR * 4) + q * 16);
            size_t elem = (size_t)tileBase * GMM_D + (size_t)i * 4;
            if (elem + 4 > totalElems) elem = 0;   // clamp (data unused)
            const float* g = x + elem;
            asm volatile("global_load_async_to_lds_b128 %0, %1, off"
                         :: "v"(lds_off), "v"(g) : "memory");
        }
        asm volatile("s_wait_asynccnt 0" ::: "memory");
    }
    __syncthreads();

    const int wave  = t >> 5;          // 0..7
    const int lane  = t & 31;
    const int k     = lane & 15;       // component (B/D column), also A row sel
    const int half  = lane >> 4;       // 0 or 1
    const int pbase = wave * 16;       // wave's 16-point sub-tile

    // ---- B-matrix registers: chunk j covers features of d=2j (rows 0,1)
    //      and d=2j+1 (rows 2,3); lane half selects which d it supplies. ----
    float wB[16], mB[16];
    #pragma unroll
    for (int j = 0; j < 16; ++j) {
        int dsel = 2 * j + half;
        float sv = sP[k * XSTR + dsel];
        float cv = cP[k * XSTR + dsel];
        wB[j] = sv * sv;               // w  = 1/sd^2   (x^2 feature weight)
        mB[j] = 2.0f * cv * sv;        // -2*mu*w       (x feature weight)
    }

    // ---- Clamp correction, accumulated NEGATIVELY into the future C ------
    // acc[r] <-> (point pbase + 8*half + r, component k): matches D layout.
    float acc[8] = {0, 0, 0, 0, 0, 0, 0, 0};
    #pragma unroll
    for (int dc = 0; dc < 8; ++dc) {
        const int d = dc * 4;
        const float4 s4 = *(const float4*)&sP[k * XSTR + d];
        const float4 c4 = *(const float4*)&cP[k * XSTR + d];
        #pragma unroll
        for (int r = 0; r < 8; ++r) {
            const int p = pbase + half * 8 + r;
            const float4 x4 = *(const float4*)&xs[p * XSTR + d]; // broadcast
            float u, v;
            u = __builtin_fmaf(x4.x, s4.x, c4.x);
            v = __builtin_fmaf(u, u, -20.0f); acc[r] -= fmaxf(v, 0.0f);
            u = __builtin_fmaf(x4.y, s4.y, c4.y);
            v = __builtin_fmaf(u, u, -20.0f); acc[r] -= fmaxf(v, 0.0f);
            u = __builtin_fmaf(x4.z, s4.z, c4.z);
            v = __builtin_fmaf(u, u, -20.0f); acc[r] -= fmaxf(v, 0.0f);
            u = __builtin_fmaf(x4.w, s4.w, c4.w);
            v = __builtin_fmaf(u, u, -20.0f); acc[r] -= fmaxf(v, 0.0f);
        }
    }

    v8f accv;
    #pragma unroll
    for (int r = 0; r < 8; ++r) accv[r] = acc[r];

    // ---- 16 chained V_WMMA_F32_16X16X4_F32: += sum_d (x^2*w - 2*mu*w*x) --
    // A (16x4): lane L = point pbase + (L%16); VGPR0 = x^2 (K=0/2),
    //           VGPR1 = x (K=1/3); lane half selects d = 2j / 2j+1.
    #pragma unroll
    for (int j = 0; j < 16; ++j) {
        float xv = xs[(pbase + k) * XSTR + 2 * j + half];
        v2f a, b;
        a.x = xv * xv;   // feature K even: x^2
        a.y = xv;        // feature K odd : x
        b.x = wB[j];     // B rows 0/2
        b.y = mB[j];     // B rows 1/3
        accv = __builtin_amdgcn_wmma_f32_16x16x4_f32(
                   false, a, false, b, (short)0, accv, false, false);
    }

    // ---- Epilogue: D vgpr r -> (point pbase+8*half+r, k) -----------------
    const float o2  = off2S[k];
    const int   gn0 = tileBase + pbase + half * 8;
    #pragma unroll
    for (int r = 0; r < 8; ++r) {
        int n = gn0 + r;
        if (n < N)
            out[(size_t)n * GMM_K + k] = __builtin_fmaf(-0.5f, accv[r], o2);
    }
}

extern "C" void kernel_launch(void* const* d_in, const int* in_sizes, int n_in,
                              void* d_out, int out_size, void* d_ws, size_t ws_size,
                              hipStream_t stream) {
    const float* x     = (const float*)d_in[0];
    const float* mu    = (const float*)d_in[1];
    const float* stdv  = (const float*)d_in[2];
    const float* alpha = (const float*)d_in[3];
    float* out = (float*)d_out;

    const int N = in_sizes[0] / GMM_D;

    float* s    = (float*)d_ws;               // [K*D]
    float* c    = s + GMM_K * GMM_D;          // [K*D]
    float* off2 = c + GMM_K * GMM_D;          // [K]

    gmm_prep<<<1, 512, 0, stream>>>(mu, stdv, alpha, s, c, off2);

    const int grid = (N + TILE - 1) / TILE;
    gmm_main<<<grid, TPB, 0, stream>>>(x, s, c, off2, out, N);
}